// PCEN_30812095381632
// MI455X (gfx1250) — compile-verified
//
#include <hip/hip_runtime.h>

#define EPS_ 1e-6f

constexpr int kB      = 32;
constexpr int kT      = 8192;
constexpr int kF      = 128;
constexpr int kChunks = 64;
constexpr int kL      = kT / kChunks;    // 128 timesteps per chunk
constexpr int kStep   = 32;              // rows per staged LDS tile (16 KB)
constexpr int kTiles  = kL / kStep;      // 4 tiles per chunk
constexpr int kThreads = kF;             // one thread per feature = 4 wave32s
constexpr int kCopies = (kStep * kF * 4) / (kThreads * 16); // 8 x b128 per thread

// Wait until outstanding async (global<->LDS) ops for this wave <= n.
#define WAIT_ASYNC(n) asm volatile("s_wait_asynccnt " #n ::: "memory")

// Issue one tile (kStep rows x kF floats) of async global->LDS b128 copies.
// Each of the 128 threads moves 8 x 16B; consecutive lanes hit consecutive
// 16B segments, so every wave issues contiguous 512B bursts.
__device__ __forceinline__ void async_tile_to_lds(const float* __restrict__ gsrc,
                                                  float* lds_dst, int tid) {
  unsigned lbase = (unsigned)(size_t)lds_dst;   // low 32 bits of generic ptr = LDS offset
  const char* g = (const char*)gsrc;
#pragma unroll
  for (int j = 0; j < kCopies; ++j) {
    unsigned off   = (unsigned)((j * kThreads + tid) * 16);
    unsigned laddr = lbase + off;
    const void* gaddr = (const void*)(g + off);
    asm volatile("global_load_async_to_lds_b128 %0, %1, off"
                 :: "v"(laddr), "v"(gaddr)
                 : "memory");
  }
}

// Passes 1 and 3 share this skeleton.
//  FINAL=false: local chunk reduce  -> Z[b][c][f] = s * sum_k a^(L-1-k) x_k
//  FINAL=true : exact chunk scan from carry_in, emit y
template <bool FINAL>
__global__ __launch_bounds__(kThreads) void pcen_chunk_kernel(
    const float* __restrict__ x,
    const float* __restrict__ log_s,
    const float* __restrict__ log_alpha,
    const float* __restrict__ log_delta,
    const float* __restrict__ log_r,
    const float* __restrict__ carry_in,   // [B][CHUNKS][F], FINAL only
    float* __restrict__ Z,                // [B][CHUNKS][F], pass-1 only
    float* __restrict__ y)                // [B][T][F],      FINAL only
{
  __shared__ __align__(16) float smem[2][kStep * kF];

  const int f = threadIdx.x;       // feature
  const int c = blockIdx.x;        // chunk
  const int b = blockIdx.y;        // batch

  const float s = __expf(log_s[f]);
  const float a = 1.0f - s;

  float alpha = 0.f, delta = 0.f, r = 0.f, dr = 0.f;
  if (FINAL) {
    alpha = __expf(log_alpha[f]);
    delta = __expf(log_delta[f]);
    r     = __expf(log_r[f]);
    dr    = __expf(r * __logf(delta));   // delta^r
  }

  float E = FINAL ? carry_in[((size_t)b * kChunks + c) * kF + f] : 0.0f;

  const float* xc = x + ((size_t)b * kT + (size_t)c * kL) * kF;
  float*       yc = FINAL ? (y + ((size_t)b * kT + (size_t)c * kL) * kF) : nullptr;

  // Prime the 2-deep async pipeline.
  async_tile_to_lds(xc,               &smem[0][0], f);
  async_tile_to_lds(xc + kStep * kF,  &smem[1][0], f);

  for (int tile = 0; tile < kTiles; ++tile) {
    if (tile == kTiles - 1) { WAIT_ASYNC(0); } else { WAIT_ASYNC(8); }
    __syncthreads();                          // all 4 waves' segments landed

    const float* buf = &smem[tile & 1][0];
#pragma unroll 4
    for (int rr = 0; rr < kStep; ++rr) {
      const float xv = buf[rr * kF + f];      // bank-conflict-free (lane==bank)
      E = fmaf(a, E, s * xv);                 // E = (1-s)E + s*x
      if (FINAL) {
        const float p    = __expf(-alpha * __logf(EPS_ + E)); // (eps+E)^-alpha
        const float base = fmaf(xv, p, delta);
        const float yv   = __expf(r * __logf(base)) - dr;
        // NT store: y is write-once / never re-read.  Keeps the 134 MB of y
        // from evicting x out of the 192 MB L2 (x fits; pass 3 re-reads it).
        __builtin_nontemporal_store(yv, &yc[(size_t)(tile * kStep + rr) * kF + f]);
      }
    }
    __syncthreads();                          // everyone done reading this buffer

    if (tile + 2 < kTiles)                    // refill the buffer just consumed
      async_tile_to_lds(xc + (size_t)(tile + 2) * kStep * kF, &smem[tile & 1][0], f);
  }

  if (!FINAL)
    Z[((size_t)b * kChunks + c) * kF + f] = E;
}

// Pass 2: serial carry chain across chunks (tiny: 4096 threads x 64 steps).
__global__ __launch_bounds__(kF) void pcen_carry_kernel(
    const float* __restrict__ state,      // [B][1][F]
    const float* __restrict__ log_s,
    const float* __restrict__ Z,          // [B][CHUNKS][F]
    float* __restrict__ carry_in,         // [B][CHUNKS][F] out
    float* __restrict__ new_state)        // [B][F] out
{
  const int f = threadIdx.x;
  const int b = blockIdx.x;
  const float s  = __expf(log_s[f]);
  const float a  = 1.0f - s;                         // in (0,1), > 0
  const float aL = __expf((float)kL * __logf(a));    // a^L via native log/exp

  float C = state[(size_t)b * kF + f];    // E entering chunk 0
  for (int c = 0; c < kChunks; ++c) {
    const size_t idx = ((size_t)b * kChunks + c) * kF + f;
    carry_in[idx] = C;
    C = fmaf(aL, C, Z[idx]);              // E at the end of chunk c
  }
  new_state[(size_t)b * kF + f] = C;      // E_{T-1}
}

extern "C" void kernel_launch(void* const* d_in, const int* in_sizes, int n_in,
                              void* d_out, int out_size, void* d_ws, size_t ws_size,
                              hipStream_t stream) {
  (void)in_sizes; (void)n_in; (void)out_size; (void)ws_size;

  const float* x         = (const float*)d_in[0];
  const float* state     = (const float*)d_in[1];
  const float* log_s     = (const float*)d_in[2];
  const float* log_alpha = (const float*)d_in[3];
  const float* log_delta = (const float*)d_in[4];
  const float* log_r     = (const float*)d_in[5];

  float* y         = (float*)d_out;                       // [B][T][F]
  float* new_state = y + (size_t)kB * kT * kF;            // [B][1][F] appended

  float* Z     = (float*)d_ws;                            // [B][CHUNKS][F]
  float* carry = Z + (size_t)kB * kChunks * kF;           // [B][CHUNKS][F]

  dim3 grid(kChunks, kB);

  // Pass 1: per-chunk local reduction (reads x once, warms L2 with x).
  pcen_chunk_kernel<false><<<grid, kThreads, 0, stream>>>(
      x, log_s, log_alpha, log_delta, log_r, nullptr, Z, nullptr);

  // Pass 2: carry propagation across chunks + final state.
  pcen_carry_kernel<<<kB, kF, 0, stream>>>(state, log_s, Z, carry, new_state);

  // Pass 3: exact chunk scan from carries, emit y (x mostly from L2, y NT-stored).
  pcen_chunk_kernel<true><<<grid, kThreads, 0, stream>>>(
      x, log_s, log_alpha, log_delta, log_r, carry, nullptr, y);
}